// NMF_2207613190802
// MI455X (gfx1250) — compile-verified
//
#include <hip/hip_runtime.h>
#include <cmath>

typedef __attribute__((ext_vector_type(2))) float v2f;
typedef __attribute__((ext_vector_type(8))) float v8f;

#define NM 4096
#define ND 8192
#define NK 64
#define BT 64           // block tile edge (M and N)
#define LDSS 68         // padded LDS row stride (floats), multiple of 4, != 64
#define W_ELEMS (NM * NK)      // 262144
#define H_ELEMS (NK * ND)      // 524288

// ---------------------------------------------------------------------------
// Fused per-iteration kernel: for one 64x64 tile of the residual grid,
//   R = W@H - md            (WMMA f32 16x16x4, K=64)
//   Gw += R @ H_tile^T      (WMMA, partials via global f32 atomics)
//   Gh += W_tile^T @ R      (WMMA, partials via global f32 atomics)
//   sumsq += ||R_tile||^2   (LDS tree reduce + one atomic per block)
// 256 threads = 8 wave32; each wave owns 2 of the 16 16x16 subtiles per phase.
// ---------------------------------------------------------------------------
__global__ __launch_bounds__(256) void nmf_fused(
    const float* __restrict__ md,
    const float* __restrict__ W,
    const float* __restrict__ H,
    float* __restrict__ Gw,
    float* __restrict__ Gh,
    float* __restrict__ sumsq,
    int slot)
{
    __shared__ float Ws[BT * LDSS];   // W tile  [m][k]
    __shared__ float Hs[BT * LDSS];   // H tile  [k][n]
    __shared__ float Rs[BT * LDSS];   // R tile  [m][n]

    const int tid  = threadIdx.x;
    const int lane = tid & 31;
    const int wave = tid >> 5;
    const int m0 = blockIdx.y * BT;
    const int n0 = blockIdx.x * BT;

    // ---- stage W (64 rows x full K=64) and H (K=64 x 64 cols) via b128 loads
    for (int i = 0; i < 4; ++i) {
        int e   = i * 1024 + tid * 4;
        int row = e >> 6;
        int col = e & 63;
        float4 w4 = *(const float4*)(W + (size_t)(m0 + row) * NK + col);
        *(float4*)(Ws + row * LDSS + col) = w4;
        float4 h4 = *(const float4*)(H + (size_t)row * ND + (n0 + col));
        *(float4*)(Hs + row * LDSS + col) = h4;
    }
    __syncthreads();

    // fragment geometry (wave32, V_WMMA_F32_16X16X4_F32)
    const int fr      = lane & 15;          // row (A) / col (B,D) within 16
    const int kb      = (lane >> 4) << 1;   // K sub-offset: 0 or 2
    const int rowBase = (lane >> 4) << 3;   // D row base: 0 or 8

    float ss = 0.0f;

    // ================= phase 1: R = W@H - md =================
    for (int it = 0; it < 2; ++it) {
        int sub = wave * 2 + it;
        int sm = (sub >> 2) << 4;
        int sn = (sub & 3) << 4;
        v8f acc = {};
        for (int k0 = 0; k0 < NK; k0 += 4) {
            v2f a, b;
            a.x = Ws[(sm + fr) * LDSS + (k0 + kb)];
            a.y = Ws[(sm + fr) * LDSS + (k0 + kb + 1)];
            b.x = Hs[(k0 + kb) * LDSS + (sn + fr)];
            b.y = Hs[(k0 + kb + 1) * LDSS + (sn + fr)];
            acc = __builtin_amdgcn_wmma_f32_16x16x4_f32(
                false, a, false, b, (short)0, acc, false, false);
        }
        for (int v = 0; v < 8; ++v) {
            int row = rowBase + v;
            float r = acc[v] -
                md[(size_t)(m0 + sm + row) * ND + (n0 + sn + fr)];
            ss += r * r;
            Rs[(sm + row) * LDSS + (sn + fr)] = r;
        }
    }
    __syncthreads();

    // ================= phase 2: Gw_partial = R @ Hs^T (64x64, K=64) =========
    // Gw[m][j] = sum_n R[m][n] * Hs[j][n]
    for (int it = 0; it < 2; ++it) {
        int sub = wave * 2 + it;
        int sm = (sub >> 2) << 4;
        int sj = (sub & 3) << 4;
        v8f acc = {};
        for (int k0 = 0; k0 < BT; k0 += 4) {
            v2f a, b;
            a.x = Rs[(sm + fr) * LDSS + (k0 + kb)];
            a.y = Rs[(sm + fr) * LDSS + (k0 + kb + 1)];
            b.x = Hs[(sj + fr) * LDSS + (k0 + kb)];       // Hs^T: row j, col n'
            b.y = Hs[(sj + fr) * LDSS + (k0 + kb + 1)];
            acc = __builtin_amdgcn_wmma_f32_16x16x4_f32(
                false, a, false, b, (short)0, acc, false, false);
        }
        for (int v = 0; v < 8; ++v) {
            int row = rowBase + v;
            atomicAdd(Gw + (size_t)(m0 + sm + row) * NK + (sj + fr), acc[v]);
        }
    }

    // ================= phase 3: Gh_partial = Ws^T @ R (64x64, K=64) =========
    // Gh[j][n] = sum_m Ws[m][j] * R[m][n]
    for (int it = 0; it < 2; ++it) {
        int sub = wave * 2 + it;
        int sj = (sub >> 2) << 4;
        int sn = (sub & 3) << 4;
        v8f acc = {};
        for (int k0 = 0; k0 < BT; k0 += 4) {
            v2f a, b;
            a.x = Ws[(k0 + kb) * LDSS + (sj + fr)];       // Ws^T: row j, col m'
            a.y = Ws[(k0 + kb + 1) * LDSS + (sj + fr)];
            b.x = Rs[(k0 + kb) * LDSS + (sn + fr)];
            b.y = Rs[(k0 + kb + 1) * LDSS + (sn + fr)];
            acc = __builtin_amdgcn_wmma_f32_16x16x4_f32(
                false, a, false, b, (short)0, acc, false, false);
        }
        for (int v = 0; v < 8; ++v) {
            int row = rowBase + v;
            atomicAdd(Gh + (size_t)(sj + row) * ND + (n0 + sn + fr), acc[v]);
        }
    }

    // ================= sumsq reduction (reuse Ws after phase 3) =============
    __syncthreads();
    Ws[tid] = ss;
    __syncthreads();
    for (int off = 128; off > 0; off >>= 1) {
        if (tid < off) Ws[tid] += Ws[tid + off];
        __syncthreads();
    }
    if (tid == 0) atomicAdd(&sumsq[slot], Ws[0]);
}

// ---------------------------------------------------------------------------
// Adam update + nonneg clamp; scales raw grads by 1/||R||_F; zeroes grad
// buffers and the *other* sumsq slot for the next iteration. On the final
// iteration also streams W||H into d_out.
// ---------------------------------------------------------------------------
__global__ __launch_bounds__(256) void adam_step(
    float* __restrict__ W,  float* __restrict__ H,
    float* __restrict__ mW, float* __restrict__ vW,
    float* __restrict__ mH, float* __restrict__ vH,
    float* __restrict__ Gw, float* __restrict__ Gh,
    float* __restrict__ sumsq, int slot,
    float ibc1, float ibc2,           // 1/(1-b1^t), 1/(1-b2^t)
    float* __restrict__ out)          // d_out on last iter, else nullptr
{
    const float LR = 1e-3f, B1 = 0.9f, B2 = 0.999f, EPS = 1e-8f;
    int i = blockIdx.x * blockDim.x + threadIdx.x;
    float inv_norm = __frsqrt_rn(sumsq[slot]);   // 1/sqrt(sum r^2) = 1/||R||
    if (i == 0) sumsq[slot ^ 1] = 0.0f;

    if (i < W_ELEMS) {
        float g = Gw[i] * inv_norm;  Gw[i] = 0.0f;
        float m = B1 * mW[i] + (1.0f - B1) * g;
        float v = B2 * vW[i] + (1.0f - B2) * g * g;
        mW[i] = m; vW[i] = v;
        float p = W[i] - LR * (m * ibc1) / (sqrtf(v * ibc2) + EPS);
        p = fmaxf(p, 0.0f);
        W[i] = p;
        if (out) out[i] = p;
    } else if (i < W_ELEMS + H_ELEMS) {
        int j = i - W_ELEMS;
        float g = Gh[j] * inv_norm;  Gh[j] = 0.0f;
        float m = B1 * mH[j] + (1.0f - B1) * g;
        float v = B2 * vH[j] + (1.0f - B2) * g * g;
        mH[j] = m; vH[j] = v;
        float p = H[j] - LR * (m * ibc1) / (sqrtf(v * ibc2) + EPS);
        p = fmaxf(p, 0.0f);
        H[j] = p;
        if (out) out[W_ELEMS + j] = p;
    }
}

__global__ __launch_bounds__(256) void nmf_init(
    const float* __restrict__ w0, const float* __restrict__ h0,
    float* __restrict__ W,  float* __restrict__ H,
    float* __restrict__ mW, float* __restrict__ vW,
    float* __restrict__ mH, float* __restrict__ vH,
    float* __restrict__ Gw, float* __restrict__ Gh,
    float* __restrict__ sumsq)
{
    int i = blockIdx.x * blockDim.x + threadIdx.x;
    if (i < 2) sumsq[i] = 0.0f;
    if (i < W_ELEMS) {
        W[i] = w0[i];
        mW[i] = 0.0f; vW[i] = 0.0f; Gw[i] = 0.0f;
    } else if (i < W_ELEMS + H_ELEMS) {
        int j = i - W_ELEMS;
        H[j] = h0[j];
        mH[j] = 0.0f; vH[j] = 0.0f; Gh[j] = 0.0f;
    }
}

extern "C" void kernel_launch(void* const* d_in, const int* in_sizes, int n_in,
                              void* d_out, int out_size, void* d_ws, size_t ws_size,
                              hipStream_t stream) {
    const float* md = (const float*)d_in[0];
    const float* w0 = (const float*)d_in[1];
    const float* h0 = (const float*)d_in[2];

    // workspace layout (floats)
    float* ws = (float*)d_ws;
    float* W     = ws;
    float* H     = W  + W_ELEMS;
    float* mW    = H  + H_ELEMS;
    float* vW    = mW + W_ELEMS;
    float* mH    = vW + W_ELEMS;
    float* vH    = mH + H_ELEMS;
    float* Gw    = vH + H_ELEMS;
    float* Gh    = Gw + W_ELEMS;
    float* sumsq = Gh + H_ELEMS;   // 2 ping-pong slots

    const int total = W_ELEMS + H_ELEMS;           // 786432
    nmf_init<<<(total + 255) / 256, 256, 0, stream>>>(
        w0, h0, W, H, mW, vW, mH, vH, Gw, Gh, sumsq);

    dim3 grid(ND / BT, NM / BT);                   // 128 x 64 tiles
    for (int t = 1; t <= 50; ++t) {
        int slot = t & 1;
        nmf_fused<<<grid, 256, 0, stream>>>(md, W, H, Gw, Gh, sumsq, slot);

        double b1t = pow(0.9, (double)t);
        double b2t = pow(0.999, (double)t);
        float ibc1 = (float)(1.0 / (1.0 - b1t));
        float ibc2 = (float)(1.0 / (1.0 - b2t));
        float* out = (t == 50) ? (float*)d_out : nullptr;
        adam_step<<<(total + 255) / 256, 256, 0, stream>>>(
            W, H, mW, vW, mH, vH, Gw, Gh, sumsq, slot, ibc1, ibc2, out);
    }
}